// YukawaComputer_83519934038695
// MI455X (gfx1250) — compile-verified
//
#include <hip/hip_runtime.h>
#include <math.h>

// ---------------------------------------------------------------------------
// YukawaComputer for MI455X (gfx1250, wave32).
// S[(i,j),k] = sum_{n,c} D[n,(i,j),c] * (Phi*w)[n,k,c] is a
// M=441 x N=77 x K=4096*35 f32 GEMM -> V_WMMA_F32_16X16X4_F32, fed by
// double-buffered GLOBAL_LOAD_ASYNC_TO_LDS (ASYNCcnt) staging.
// ---------------------------------------------------------------------------

typedef float v2f __attribute__((ext_vector_type(2)));
typedef float v8f __attribute__((ext_vector_type(8)));

#define NH3        77
#define NTRI       35
#define NROWS      441            // 21*21
#define ROWS_PAD   448            // 28 M-tiles of 16
#define COLS_PAD   80             // 5 N-tiles of 16
#define CPAD       36             // c padded to 36 = 9 k-steps of 4
#define NPB        16             // batch samples per block
#define TW         320            // 10 wave32s per block
#define PHI_N      2695           // 77*35 floats per sample
#define PHI_PAD    2816           // room for k=77..79 garbage reads
#define TFIN       512

// ---- constexpr generation of the sparse wedge structure constants ---------
struct WedgeTab { short a[NTRI][6]; short b[NTRI][6]; float s[NTRI][6]; };

constexpr int pair_idx(int x, int y) { return x * (13 - x) / 2 + (y - x - 1); }

constexpr WedgeTab make_tab() {
  WedgeTab t{};
  int c = 0;
  for (int i = 0; i < 7; i++)
    for (int j = i + 1; j < 7; j++)
      for (int k = j + 1; k < 7; k++) {
        int rem[4] = {0, 0, 0, 0};
        int m = 0;
        for (int x = 0; x < 7; x++)
          if (x != i && x != j && x != k) rem[m++] = x;
        const int sp[3][4] = {{0, 1, 2, 3}, {0, 2, 1, 3}, {0, 3, 1, 2}};
        int e = 0;
        for (int u = 0; u < 3; u++) {
          int a0 = rem[sp[u][0]], a1 = rem[sp[u][1]];
          int b0 = rem[sp[u][2]], b1 = rem[sp[u][3]];
          int perm[7] = {a0, a1, b0, b1, i, j, k};
          int inv = 0;
          for (int p = 0; p < 7; p++)
            for (int q = p + 1; q < 7; q++)
              if (perm[p] > perm[q]) inv++;
          float sg = (inv & 1) ? -1.f : 1.f;
          t.a[c][e] = (short)pair_idx(a0, a1);
          t.b[c][e] = (short)pair_idx(b0, b1);
          t.s[c][e] = sg; e++;
          t.a[c][e] = (short)pair_idx(b0, b1);
          t.b[c][e] = (short)pair_idx(a0, a1);
          t.s[c][e] = sg; e++;
        }
        c++;
      }
  return t;
}
static constexpr WedgeTab TAB = make_tab();

// ---- CDNA5 async global->LDS helpers (ASYNCcnt path) ----------------------
// Low 32 bits of a flat pointer into the LDS aperture are the LDS byte
// address (ISA 10.2: aperture base has zero low bits).
__device__ __forceinline__ unsigned lds_off(const void* p) {
  return (unsigned)(uintptr_t)p;
}
__device__ __forceinline__ void async_b128(unsigned lds, const float* g) {
  asm volatile("global_load_async_to_lds_b128 %0, %1, off"
               :: "v"(lds), "v"(g) : "memory");
}
__device__ __forceinline__ void async_b32(unsigned lds, const float* g) {
  asm volatile("global_load_async_to_lds_b32 %0, %1, off"
               :: "v"(lds), "v"(g) : "memory");
}
__device__ __forceinline__ void wait_async0() {
  asm volatile("s_wait_asynccnt 0x0" ::: "memory");
}

// ---- kernel 1: vol[n] = sqrt(|det(metric_n)|), plus global sum ------------
__global__ __launch_bounds__(256)
void vol_kernel(const float* __restrict__ metric, float* __restrict__ vol,
                float* __restrict__ vsum, int batch) {
  const int n = blockIdx.x * 256 + threadIdx.x;
  float v = 0.f;
  if (n < batch) {
    float a[7][7];
    const float* m = metric + (size_t)n * 49;
    for (int r = 0; r < 7; r++)
      for (int c = 0; c < 7; c++) a[r][c] = m[r * 7 + c];
    float det = 1.f;
    for (int k = 0; k < 7; k++) {
      int piv = k; float mx = fabsf(a[k][k]);
      for (int r = k + 1; r < 7; r++) {
        float t = fabsf(a[r][k]);
        if (t > mx) { mx = t; piv = r; }
      }
      if (piv != k) {
        for (int c = k; c < 7; c++) { float t = a[k][c]; a[k][c] = a[piv][c]; a[piv][c] = t; }
        det = -det;
      }
      const float akk = a[k][k];
      det *= akk;
      const float invp = (akk != 0.f) ? 1.f / akk : 0.f;
      for (int r = k + 1; r < 7; r++) {
        const float f = a[r][k] * invp;
        for (int c = k; c < 7; c++) a[r][c] = fmaf(-f, a[k][c], a[r][c]);
      }
    }
    v = sqrtf(fabsf(det));
    vol[n] = v;
  }
  __shared__ float red[256];
  red[threadIdx.x] = v;
  __syncthreads();
  for (int s0 = 128; s0 > 0; s0 >>= 1) {
    if (threadIdx.x < s0) red[threadIdx.x] += red[threadIdx.x + s0];
    __syncthreads();
  }
  if (threadIdx.x == 0) atomicAdd(vsum, red[0]);
}

// ---- kernel 2: WMMA contraction with async double-buffered staging --------
__global__ __launch_bounds__(TW)
void wedge_kernel(const float* __restrict__ omega, const float* __restrict__ Phi,
                  const float* __restrict__ vol, float* __restrict__ Sws, int batch) {
  __shared__ float PhiB[2][PHI_PAD];       // raw Phi[n], [k][35] layout
  __shared__ float OmB[2][448];            // raw omega[n]
  __shared__ float Dl[224 * CPAD];         // half of D panel (224 x 36), vn folded in

  const int tid   = threadIdx.x;
  const int wave  = tid >> 5;
  const int lane  = tid & 31;
  const int ntile = wave % 5;
  const int mpar  = wave / 5;
  const int lhalf = lane >> 4;
  const int lidx  = lane & 15;

  const unsigned pb0 = lds_off(&PhiB[0][0]), pb1 = lds_off(&PhiB[1][0]);
  const unsigned ob0 = lds_off(&OmB[0][0]),  ob1 = lds_off(&OmB[1][0]);

  v8f acc[14];
  #pragma unroll
  for (int s0 = 0; s0 < 14; s0++)
    #pragma unroll
    for (int e = 0; e < 8; e++) acc[s0][e] = 0.f;

  const int n0 = blockIdx.x * NPB;
  const int nmax = (batch - n0 < NPB) ? (batch - n0) : NPB;

  // prologue: kick off sample 0 staging
  {
    const float* phin = Phi + (size_t)n0 * PHI_N;
    const float* omn  = omega + (size_t)n0 * NROWS;
    for (int c0 = tid; c0 < 673; c0 += TW) async_b128(pb0 + 16u * c0, phin + 4 * c0);
    for (int c0 = tid; c0 < 110; c0 += TW) async_b128(ob0 + 16u * c0, omn + 4 * c0);
    if (tid < 3)        async_b32(pb0 + 4u * (2692 + tid), phin + 2692 + tid);
    else if (tid == 3)  async_b32(ob0 + 4u * 440, omn + 440);
  }

  for (int nn = 0; nn < nmax; ++nn) {
    const int n = n0 + nn;
    const int cur = nn & 1;
    wait_async0();               // this wave's async writes for sample nn done
    __syncthreads();             // all waves' writes visible; prev buffers free
    if (nn + 1 < nmax) {         // prefetch next sample into the other buffer
      const unsigned pb = cur ? pb0 : pb1;
      const unsigned ob = cur ? ob0 : ob1;
      const float* phin = Phi + (size_t)(n + 1) * PHI_N;
      const float* omn  = omega + (size_t)(n + 1) * NROWS;
      for (int c0 = tid; c0 < 673; c0 += TW) async_b128(pb + 16u * c0, phin + 4 * c0);
      for (int c0 = tid; c0 < 110; c0 += TW) async_b128(ob + 16u * c0, omn + 4 * c0);
      if (tid < 3)        async_b32(pb + 4u * (2692 + tid), phin + 2692 + tid);
      else if (tid == 3)  async_b32(ob + 4u * 440, omn + 440);
    }
    const float vn = vol[n];
    const float* om = &OmB[cur][0];
    const float* pc = &PhiB[cur][0];

    #pragma unroll
    for (int hh = 0; hh < 2; ++hh) {
      __syncthreads();           // Dl consumers from previous half done
      // D panel rows [hh*224, hh*224+224), weight vn folded in; c=35 pad = 0
      // (the zero A-column makes the un-padded raw-Phi B panel safe at K=35).
      for (int t = tid; t < 224 * CPAD; t += TW) {
        const int lr = t / CPAD, c = t - lr * CPAD;
        const int grow = hh * 224 + lr;
        float v = 0.f;
        if (c < NTRI && grow < NROWS) {
          const int i = grow / 21, j = grow - i * 21;
          const float* oi = om + i * 21;
          const float* oj = om + j * 21;
          #pragma unroll
          for (int e = 0; e < 6; e++)
            v = fmaf(TAB.s[c][e], oi[TAB.a[c][e]] * oj[TAB.b[c][e]], v);
          v *= vn;
        }
        Dl[t] = v;
      }
      __syncthreads();
      // 9 K-steps of V_WMMA_F32_16X16X4_F32; B-frag reused across 7 M-tiles.
      const int bcol = ntile * 16 + lidx;        // output column (k of Phi)
      const int brow = bcol * NTRI;              // raw Phi row base
      for (int ks = 0; ks < 9; ++ks) {
        const int kb = 4 * ks + 2 * lhalf;       // lanes 0-15: K{0,1}; 16-31: K{2,3}
        v2f bf;
        bf.x = pc[brow + kb];
        bf.y = pc[brow + kb + 1];                // K=35 slot zeroed on A side
        #pragma unroll
        for (int q = 0; q < 7; q++) {
          const int lrow = (2 * q + mpar) * 16 + lidx;
          v2f af;
          af.x = Dl[lrow * CPAD + kb];
          af.y = Dl[lrow * CPAD + kb + 1];
          acc[hh * 7 + q] = __builtin_amdgcn_wmma_f32_16x16x4_f32(
              false, af, false, bf, (short)0, acc[hh * 7 + q], false, false);
        }
      }
    }
  }
  // Flush: C/D layout -> row = e + 8*(lane>>4), col = lane&15 within tile.
  #pragma unroll
  for (int slot = 0; slot < 14; ++slot) {
    const int hh = slot / 7, q = slot - hh * 7;
    const int mt = hh * 14 + 2 * q + mpar;
    #pragma unroll
    for (int e = 0; e < 8; e++) {
      const int row = mt * 16 + e + 8 * lhalf;
      const int col = ntile * 16 + lidx;
      atomicAdd(&Sws[row * COLS_PAD + col], acc[slot][e]);
    }
  }
}

// ---- kernel 3: Y, M = inv^2 * S^T S, Jacobi eigenvalues -------------------
#define ALD 79
__global__ __launch_bounds__(TFIN)
void finalize_kernel(const float* __restrict__ Sws, const float* __restrict__ vsum,
                     float* __restrict__ out) {
  __shared__ float Amat[77 * ALD];
  __shared__ int   rp[39], rq[39];
  __shared__ float rc[39], rs[39];
  const int tid = threadIdx.x;
  const float inv = 1.f / vsum[0];
  const float inv2 = inv * inv;

  for (int idx = tid; idx < NROWS * NH3; idx += TFIN) {
    const int r = idx / NH3, k = idx - r * NH3;
    const int i = r / 21, j = r - i * 21;
    const float v = Sws[r * COLS_PAD + k] * inv;
    out[idx] = (j >= i) ? v : -v;
  }
  for (int idx = tid; idx < NH3 * NH3; idx += TFIN) {
    const int k = idx / NH3, l = idx - k * NH3;
    float a = 0.f;
    for (int r = 0; r < NROWS; ++r)
      a = fmaf(Sws[r * COLS_PAD + k], Sws[r * COLS_PAD + l], a);
    a *= inv2;
    Amat[k * ALD + l] = a;
    out[NROWS * NH3 + idx] = a;
  }
  __syncthreads();

  for (int sweep = 0; sweep < 10; ++sweep) {
    for (int round = 0; round < 77; ++round) {
      if (tid < 39) {
        const int posA = tid, posB = 77 - tid;
        int p = (posA == 0) ? 0 : 1 + (posA - 1 + round) % 77;
        int q = 1 + (posB - 1 + round) % 77;
        if (p == 77 || q == 77) {
          rp[tid] = 0; rq[tid] = 0; rc[tid] = 1.f; rs[tid] = 0.f;
        } else {
          if (p > q) { const int t = p; p = q; q = t; }
          const float app = Amat[p * ALD + p], aqq = Amat[q * ALD + q];
          const float apq = Amat[p * ALD + q];
          float c = 1.f, s = 0.f;
          if (apq != 0.f) {
            const float th = (aqq - app) / (2.f * apq);
            const float tt = ((th >= 0.f) ? 1.f : -1.f) /
                             (fabsf(th) + sqrtf(th * th + 1.f));
            c = 1.f / sqrtf(tt * tt + 1.f);
            s = tt * c;
          }
          rp[tid] = p; rq[tid] = q; rc[tid] = c; rs[tid] = s;
        }
      }
      __syncthreads();
      for (int idx = tid; idx < 39 * 77; idx += TFIN) {    // A <- A * J
        const int t = idx / 77, rr = idx - t * 77;
        const int p = rp[t], q = rq[t];
        if (p == q) continue;
        const float c = rc[t], s = rs[t];
        const float x = Amat[rr * ALD + p], y = Amat[rr * ALD + q];
        Amat[rr * ALD + p] = c * x - s * y;
        Amat[rr * ALD + q] = s * x + c * y;
      }
      __syncthreads();
      for (int idx = tid; idx < 39 * 77; idx += TFIN) {    // A <- J^T * A
        const int t = idx / 77, cc = idx - t * 77;
        const int p = rp[t], q = rq[t];
        if (p == q) continue;
        const float c = rc[t], s = rs[t];
        const float x = Amat[p * ALD + cc], y = Amat[q * ALD + cc];
        Amat[p * ALD + cc] = c * x - s * y;
        Amat[q * ALD + cc] = s * x + c * y;
      }
      __syncthreads();
    }
  }
  if (tid == 0) {
    float ev[77];
    for (int k = 0; k < 77; k++) ev[k] = Amat[k * ALD + k];
    for (int i = 1; i < 77; i++) {
      const float x = ev[i]; int j = i - 1;
      while (j >= 0 && ev[j] < x) { ev[j + 1] = ev[j]; --j; }
      ev[j + 1] = x;
    }
    for (int k = 0; k < 77; k++) out[NROWS * NH3 + NH3 * NH3 + k] = ev[k];
  }
}

// ---------------------------------------------------------------------------
extern "C" void kernel_launch(void* const* d_in, const int* in_sizes, int n_in,
                              void* d_out, int out_size, void* d_ws, size_t ws_size,
                              hipStream_t stream) {
  const float* omega  = (const float*)d_in[0];
  const float* Phi    = (const float*)d_in[1];
  const float* metric = (const float*)d_in[2];
  float* out = (float*)d_out;
  float* ws  = (float*)d_ws;
  const int batch = in_sizes[2] / 49;

  float* wvol = ws;                 // [batch]
  float* wsum = ws + batch;         // [1] (+7 pad)
  float* wS   = ws + batch + 8;     // [448*80] padded S accumulator

  hipMemsetAsync(wsum, 0, (8 + ROWS_PAD * COLS_PAD) * sizeof(float), stream);

  vol_kernel<<<(batch + 255) / 256, 256, 0, stream>>>(metric, wvol, wsum, batch);
  const int chunks = (batch + NPB - 1) / NPB;
  wedge_kernel<<<chunks, TW, 0, stream>>>(omega, Phi, wvol, wS, batch);
  finalize_kernel<<<1, TFIN, 0, stream>>>(wS, wsum, out);
}